// NodeGCNConv_32701880992040
// MI455X (gfx1250) — compile-verified
//
#include <hip/hip_runtime.h>
#include <hip/hip_bf16.h>

typedef __attribute__((ext_vector_type(2))) float v2f;
typedef __attribute__((ext_vector_type(8))) float v8f;

#define C_E   64
#define C_N   256
#define C_OUT 256
#define KTOT  320            // 64 (A/D @ W_pass) + 256 (X @ W_self)
#define BR    48             // node rows per block (3 row-tiles of 16)
#define LDS_STRIDE 324       // 320 + pad; 324 % 64 == 4 -> conflict-free frag reads

// ---------------------------------------------------------------------------
// Stage 1: segment_sum(edge_attr, row_idx) -> A (N x 64) via L2 float atomics.
// One thread per (edge, 4-channel group): float4 load + 4 atomic adds.
// A (12.8 MB) is fully L2-resident (192 MB); the 410 MB edge stream from HBM
// is the real cost -> this stage sits at the bandwidth floor.
// ---------------------------------------------------------------------------
__global__ __launch_bounds__(256) void gcn_scatter_kernel(
    const float* __restrict__ edge_attr,
    const int* __restrict__ row_idx,
    float* __restrict__ A,
    int nEdges)
{
    int idx = blockIdx.x * 256 + threadIdx.x;
    int e = idx >> 4;            // 16 groups of 4 channels per edge
    if (e >= nEdges) return;
    int g = (idx & 15) << 2;     // channel base 0..60
    int node = row_idx[e];
    const float4 v = *(const float4*)(edge_attr + (size_t)e * C_E + g);
    float* dst = A + (size_t)node * C_E + g;
    atomicAdd(dst + 0, v.x);
    atomicAdd(dst + 1, v.y);
    atomicAdd(dst + 2, v.z);
    atomicAdd(dst + 3, v.w);
}

// ---------------------------------------------------------------------------
// Stage 2: out = ReLU( [A/D | X] @ [W_pass ; W_self] + b_pass + b_self )
// Block: 256 threads = 8 wave32, 48 node rows (3 row-tiles), 256 output cols.
// Wave w owns col tiles [32w, 32w+16) and [32w+16, 32w+32) for all 3 row
// tiles -> each B fragment fetched once per K-step feeds 3 WMMAs (3x reuse,
// cutting L2 weight traffic vs. 2-row-tile version). K accumulated with
// V_WMMA_F32_16X16X4_F32 in full fp32 (problem is bandwidth-bound; no reason
// to quantize).
// ---------------------------------------------------------------------------
__global__ __launch_bounds__(256) void gcn_fused_gemm_kernel(
    const float* __restrict__ A,      // (N, 64) from stage 1
    const float* __restrict__ D,      // (N,)
    const float* __restrict__ X,      // (N, 256)
    const float* __restrict__ Wp,     // (64, 256)
    const float* __restrict__ bp,     // (256,)
    const float* __restrict__ Ws,     // (256, 256)
    const float* __restrict__ bs,     // (256,)
    float* __restrict__ out,          // (N, 256)
    int nNodes)
{
    __shared__ float rowbuf[BR * LDS_STRIDE];   // 48 x 320 fused rows, ~62 KB
    __shared__ float rdbuf[BR];                 // 1/D per row

    const int tid  = threadIdx.x;
    const int row0 = blockIdx.x * BR;

    // reciprocal of D once per row
    if (tid < BR) {
        int node = row0 + tid;
        rdbuf[tid] = (node < nNodes) ? (1.0f / D[node]) : 0.0f;
    }
    __syncthreads();

    // Stage concat[A/D, X] rows into LDS (zero-fill ragged tail rows).
    for (int i = tid; i < BR * KTOT; i += 256) {
        int r = i / KTOT;
        int k = i - r * KTOT;
        int node = row0 + r;
        float v = 0.0f;
        if (node < nNodes) {
            if (k < C_E) v = A[(size_t)node * C_E + k] * rdbuf[r];
            else         v = X[(size_t)node * C_N + (k - C_E)];
        }
        rowbuf[r * LDS_STRIDE + k] = v;
    }
    __syncthreads();

    const int wave = tid >> 5;
    const int lane = tid & 31;
    const bool hi  = lane >= 16;
    const int  l16 = lane & 15;
    const int  koff = hi ? 2 : 0;     // K sub-pair per ISA 16x4 f32 A/B layout
    const int  cb0  = wave * 32;      // this wave's column base (two 16-col tiles)

    // Accumulators seeded with bias (same value down each column).
    const float bias0 = bp[cb0 + l16]      + bs[cb0 + l16];
    const float bias1 = bp[cb0 + 16 + l16] + bs[cb0 + 16 + l16];
    v8f c00 = {bias0, bias0, bias0, bias0, bias0, bias0, bias0, bias0};
    v8f c01 = {bias1, bias1, bias1, bias1, bias1, bias1, bias1, bias1};
    v8f c10 = c00, c20 = c00;
    v8f c11 = c01, c21 = c01;

    const float* ldsA0 = rowbuf + l16 * LDS_STRIDE;   // row tile 0
    const float* ldsA1 = ldsA0 + 16 * LDS_STRIDE;     // row tile 1
    const float* ldsA2 = ldsA0 + 32 * LDS_STRIDE;     // row tile 2

    // ---- K = 0..63 : (A/D) @ W_pass ----
    #pragma unroll 4
    for (int k0 = 0; k0 < C_E; k0 += 4) {
        const int kk = k0 + koff;
        v2f a0, a1, a2, b0, b1;
        a0.x = ldsA0[kk];  a0.y = ldsA0[kk + 1];
        a1.x = ldsA1[kk];  a1.y = ldsA1[kk + 1];
        a2.x = ldsA2[kk];  a2.y = ldsA2[kk + 1];
        const float* w0 = Wp + (size_t)kk * C_OUT;
        const float* w1 = w0 + C_OUT;
        b0.x = w0[cb0 + l16];      b0.y = w1[cb0 + l16];
        b1.x = w0[cb0 + 16 + l16]; b1.y = w1[cb0 + 16 + l16];
        c00 = __builtin_amdgcn_wmma_f32_16x16x4_f32(false, a0, false, b0, (short)0, c00, false, false);
        c01 = __builtin_amdgcn_wmma_f32_16x16x4_f32(false, a0, false, b1, (short)0, c01, false, false);
        c10 = __builtin_amdgcn_wmma_f32_16x16x4_f32(false, a1, false, b0, (short)0, c10, false, false);
        c11 = __builtin_amdgcn_wmma_f32_16x16x4_f32(false, a1, false, b1, (short)0, c11, false, false);
        c20 = __builtin_amdgcn_wmma_f32_16x16x4_f32(false, a2, false, b0, (short)0, c20, false, false);
        c21 = __builtin_amdgcn_wmma_f32_16x16x4_f32(false, a2, false, b1, (short)0, c21, false, false);
    }

    // ---- K = 64..319 : X @ W_self ----
    #pragma unroll 4
    for (int k0 = 0; k0 < C_N; k0 += 4) {
        const int kk = k0 + koff;
        v2f a0, a1, a2, b0, b1;
        a0.x = ldsA0[C_E + kk];  a0.y = ldsA0[C_E + kk + 1];
        a1.x = ldsA1[C_E + kk];  a1.y = ldsA1[C_E + kk + 1];
        a2.x = ldsA2[C_E + kk];  a2.y = ldsA2[C_E + kk + 1];
        const float* w0 = Ws + (size_t)kk * C_OUT;
        const float* w1 = w0 + C_OUT;
        b0.x = w0[cb0 + l16];      b0.y = w1[cb0 + l16];
        b1.x = w0[cb0 + 16 + l16]; b1.y = w1[cb0 + 16 + l16];
        c00 = __builtin_amdgcn_wmma_f32_16x16x4_f32(false, a0, false, b0, (short)0, c00, false, false);
        c01 = __builtin_amdgcn_wmma_f32_16x16x4_f32(false, a0, false, b1, (short)0, c01, false, false);
        c10 = __builtin_amdgcn_wmma_f32_16x16x4_f32(false, a1, false, b0, (short)0, c10, false, false);
        c11 = __builtin_amdgcn_wmma_f32_16x16x4_f32(false, a1, false, b1, (short)0, c11, false, false);
        c20 = __builtin_amdgcn_wmma_f32_16x16x4_f32(false, a2, false, b0, (short)0, c20, false, false);
        c21 = __builtin_amdgcn_wmma_f32_16x16x4_f32(false, a2, false, b1, (short)0, c21, false, false);
    }

    // ---- fused ReLU + store (D layout: VGPR r -> M = r (+8 for hi lanes)) ----
    const int mbase = hi ? 8 : 0;
    const int col0  = cb0 + l16;
    const int col1  = cb0 + 16 + l16;

    if (row0 + BR <= nNodes) {
        // full block: no per-row guards -> clean store clauses
        #pragma unroll
        for (int r = 0; r < 8; ++r) {
            const int m = r + mbase;
            float* o0 = out + (size_t)(row0 + m)      * C_OUT;
            float* o1 = out + (size_t)(row0 + 16 + m) * C_OUT;
            float* o2 = out + (size_t)(row0 + 32 + m) * C_OUT;
            o0[col0] = fmaxf(c00[r], 0.0f);  o0[col1] = fmaxf(c01[r], 0.0f);
            o1[col0] = fmaxf(c10[r], 0.0f);  o1[col1] = fmaxf(c11[r], 0.0f);
            o2[col0] = fmaxf(c20[r], 0.0f);  o2[col1] = fmaxf(c21[r], 0.0f);
        }
    } else {
        // ragged tail block (at most one per grid)
        #pragma unroll
        for (int r = 0; r < 8; ++r) {
            const int m = r + mbase;
            const int n0 = row0 + m, n1 = row0 + 16 + m, n2 = row0 + 32 + m;
            if (n0 < nNodes) {
                out[(size_t)n0 * C_OUT + col0] = fmaxf(c00[r], 0.0f);
                out[(size_t)n0 * C_OUT + col1] = fmaxf(c01[r], 0.0f);
            }
            if (n1 < nNodes) {
                out[(size_t)n1 * C_OUT + col0] = fmaxf(c10[r], 0.0f);
                out[(size_t)n1 * C_OUT + col1] = fmaxf(c11[r], 0.0f);
            }
            if (n2 < nNodes) {
                out[(size_t)n2 * C_OUT + col0] = fmaxf(c20[r], 0.0f);
                out[(size_t)n2 * C_OUT + col1] = fmaxf(c21[r], 0.0f);
            }
        }
    }
}

extern "C" void kernel_launch(void* const* d_in, const int* in_sizes, int n_in,
                              void* d_out, int out_size, void* d_ws, size_t ws_size,
                              hipStream_t stream) {
    const float* D         = (const float*)d_in[0];
    const int*   row_idx   = (const int*)  d_in[1];
    const float* edge_attr = (const float*)d_in[2];
    const float* X         = (const float*)d_in[3];
    const float* Wp        = (const float*)d_in[4];
    const float* bp        = (const float*)d_in[5];
    const float* Ws        = (const float*)d_in[6];
    const float* bs        = (const float*)d_in[7];
    float* out = (float*)d_out;

    const int nNodes = in_sizes[0];
    const int nEdges = in_sizes[1];

    float* A = (float*)d_ws;   // (nNodes, 64) scratch

    // zero the segment-sum accumulator (stream-ordered; graph-capturable)
    hipMemsetAsync(A, 0, (size_t)nNodes * C_E * sizeof(float), stream);

    // stage 1: scatter-add edge features
    {
        long long threads = (long long)nEdges * 16;
        int blocks = (int)((threads + 255) / 256);
        gcn_scatter_kernel<<<blocks, 256, 0, stream>>>(edge_attr, row_idx, A, nEdges);
    }

    // stage 2: fused normalize + dual GEMM + bias + ReLU via fp32 WMMA
    {
        int blocks = (nNodes + BR - 1) / BR;
        gcn_fused_gemm_kernel<<<blocks, 256, 0, stream>>>(A, D, X, Wp, bp, Ws, bs, out, nNodes);
    }
}